// QuaLinearScore_73315091742820
// MI455X (gfx1250) — compile-verified
//
#include <hip/hip_runtime.h>

// CDNA5 / gfx1250, wave32.
// score[n,a,b,c] = sum_k (word@U)*(neigh@T)[na,k] * (tag@V)[b,k] * (tag@W)[c,k]
// Implemented as:  P = (word@Up)*(neigh@Tp)  [16384 x 400pad]  (WMMA f32 16x16x4,
//                      U/T slabs staged via global_load_async_to_lds_b128)
//                  Gt[k][n] = g2[b,k]*g3[c,k], n=b*50+c        [400pad x 2560pad]
//                  score = P @ Gt                              (WMMA f32 16x16x4,
//                      B slabs staged via global_load_async_to_lds_b128)

typedef __attribute__((ext_vector_type(2))) float v2f;
typedef __attribute__((ext_vector_type(8))) float v8f;

#define WEMD     400
#define RANK     396
#define RANK_PAD 400      // pad rank to multiple of 16; pad cols/rows are zero
#define TAGSET   50
#define TEMD     20
#define SEQ      512
#define BATCH    32
#define M_TOTAL  (SEQ * BATCH)        // 16384
#define N_TOTAL  (TAGSET * TAGSET)    // 2500
#define N_PAD    2560                 // multiple of 32, zero-padded
#define KCHUNK   16                   // k-rows staged per LDS chunk
#define NCHUNK_S (RANK_PAD / KCHUNK)  // 25 (k_score)
#define NCHUNK_G (WEMD / KCHUNK)      // 25 (k_gates)

static __device__ __forceinline__ v8f wmma_f32(v2f a, v2f b, v8f c) {
    return __builtin_amdgcn_wmma_f32_16x16x4_f32(
        /*neg_a=*/false, a, /*neg_b=*/false, b,
        /*c_mod=*/(short)0, c, /*reuse_a=*/false, /*reuse_b=*/false);
}

// Async DMA one b128 from global to LDS. ldsOff is the byte offset within this
// workgroup's LDS allocation; escPtr is a pointer to the __shared__ buffer so
// its address escapes and the "memory" clobber forces the compiler to treat
// this asm as a writer of that buffer (keeps the ds_load consumers alive).
static __device__ __forceinline__ void async_b128(unsigned ldsOff,
                                                  const void* gaddr,
                                                  const void* escPtr) {
    asm volatile("global_load_async_to_lds_b128 %0, %1, off"
                 :: "v"(ldsOff),
                    "v"((unsigned long long)(uintptr_t)gaddr),
                    "v"(escPtr)
                 : "memory");
}
static __device__ __forceinline__ void wait_async(void) {
    asm volatile("s_wait_asynccnt 0x0" ::: "memory");
}
static __device__ __forceinline__ void wait_async1(void) {
    asm volatile("s_wait_asynccnt 0x1" ::: "memory");
}

// ---------------------------------------------------------------- g2, g3
__global__ void k_g23(const float* __restrict__ tag,
                      const float* __restrict__ V,
                      const float* __restrict__ W,
                      float* __restrict__ g2, float* __restrict__ g3) {
    int idx = blockIdx.x * blockDim.x + threadIdx.x;
    const int total = TAGSET * RANK;
    if (idx >= 2 * total) return;
    const float* Mx = (idx < total) ? V : W;
    float* out      = (idx < total) ? g2 : g3;
    int t = (idx < total) ? idx : (idx - total);
    int b = t / RANK, k = t % RANK;
    float s = 0.0f;
    #pragma unroll
    for (int e = 0; e < TEMD; ++e) s += tag[b * TEMD + e] * Mx[e * RANK + k];
    out[b * RANK + k] = s;
}

// ---------------------------------------------------- Gt[k][n] (padded, zeros)
__global__ void k_buildG(const float* __restrict__ g2,
                         const float* __restrict__ g3,
                         float* __restrict__ Gt) {
    int idx = blockIdx.x * blockDim.x + threadIdx.x;
    if (idx >= RANK_PAD * N_PAD) return;
    int k = idx / N_PAD, n = idx % N_PAD;
    float v = 0.0f;
    if (k < RANK && n < N_TOTAL) {
        int b = n / TAGSET, c = n % TAGSET;
        v = g2[b * RANK + k] * g3[c * RANK + k];
    }
    Gt[idx] = v;
}

// --------------------------- Up/Tp [400][400]: U,T with cols 396..399 zeroed
__global__ void k_padUT(const float* __restrict__ U, const float* __restrict__ T,
                        float* __restrict__ Up, float* __restrict__ Tp) {
    int idx = blockIdx.x * blockDim.x + threadIdx.x;
    const int total = WEMD * RANK_PAD;
    if (idx >= 2 * total) return;
    int sel = (idx >= total);
    int t = sel ? (idx - total) : idx;
    int r = t / RANK_PAD, c = t % RANK_PAD;
    const float* src = sel ? T : U;
    float* dst       = sel ? Tp : Up;
    dst[t] = (c < RANK) ? src[r * RANK + c] : 0.0f;
}

// ---------------- P = (word@Up)*(neigh@Tp).  Block = 4 waves = 256M x 16cols.
// Per chunk: 16x16 slab of Up AND Tp async-staged to LDS (1 b128 per thread).
__global__ __launch_bounds__(128) void k_gates(const float* __restrict__ word,
                                               const float* __restrict__ first,
                                               const float* __restrict__ Up,
                                               const float* __restrict__ Tp,
                                               float* __restrict__ P) {
    __shared__ __attribute__((aligned(16))) float sUT[2][2][KCHUNK * 16]; // 4KB

    const int tid  = threadIdx.x;
    const int lane = tid & 31;
    const int wave = tid >> 5;                    // 0..3
    const int mb   = blockIdx.x * 256 + wave * 64;
    const int rb   = blockIdx.y * 16;             // 400/16 = 25 blocks
    const int mrow = lane & 15;
    const int koff = (lane >> 4) * 2;             // 0 or 2
    const int ncol = lane & 15;

    // Staging geometry: tid -> {U|T}, row 0..15, 4-float group.
    const int sel   = tid >> 6;                   // 0=U, 1=T
    const int su    = tid & 63;
    const int srow  = su >> 2;                    // 0..15
    const int scol4 = (su & 3) * 4;               // 0,4,8,12
    const float* bsrc = sel ? Tp : Up;

    const v8f vzero = {0.f,0.f,0.f,0.f,0.f,0.f,0.f,0.f};
    v8f acc0[4], acc1[4];
    const float* arow[4];
    const float* nrow[4];
    #pragma unroll
    for (int t = 0; t < 4; ++t) {
        acc0[t] = vzero; acc1[t] = vzero;
        int m = mb + t * 16 + mrow;
        arow[t] = word + (size_t)m * WEMD;
        nrow[t] = (m >= BATCH) ? (word + (size_t)(m - BATCH) * WEMD) : first;
    }

    // sUT is the only __shared__ object -> LDS byte offset 0.
    #define ISSUE_UT(c)                                                         \
        async_b128((unsigned)(((((c) & 1) * 2 + sel) * (KCHUNK * 16) +          \
                               srow * 16 + scol4) * sizeof(float)),             \
                   bsrc + (size_t)((c) * KCHUNK + srow) * RANK_PAD + rb + scol4,\
                   &sUT[0][0][0])

    ISSUE_UT(0);
    for (int c = 0; c < NCHUNK_G; ++c) {
        if (c + 1 < NCHUNK_G) { ISSUE_UT(c + 1); wait_async1(); }
        else                  { wait_async(); }
        __syncthreads();

        #pragma unroll
        for (int kl = 0; kl < KCHUNK; kl += 4) {
            const int r0 = (kl + koff) * 16 + ncol;
            v2f bu, bt;
            bu.x = sUT[c & 1][0][r0];  bu.y = sUT[c & 1][0][r0 + 16];
            bt.x = sUT[c & 1][1][r0];  bt.y = sUT[c & 1][1][r0 + 16];
            const int kb = c * KCHUNK + kl;
            #pragma unroll
            for (int t = 0; t < 4; ++t) {
                v2f aw = *(const v2f*)(arow[t] + kb + koff);  // 8B aligned
                v2f an = *(const v2f*)(nrow[t] + kb + koff);
                acc0[t] = wmma_f32(aw, bu, acc0[t]);
                acc1[t] = wmma_f32(an, bt, acc1[t]);
            }
        }
        __syncthreads();
    }
    #undef ISSUE_UT

    // P row-major [M][RANK_PAD]; pad columns are exact zeros (Up/Tp pads).
    const int k = rb + ncol;
    #pragma unroll
    for (int t = 0; t < 4; ++t) {
        #pragma unroll
        for (int i = 0; i < 8; ++i) {
            int m = mb + t * 16 + ((lane >> 4) << 3) + i;
            P[(size_t)m * RANK_PAD + k] = acc0[t][i] * acc1[t][i];
        }
    }
}

// ------------- score = P @ Gt.  Block = 4 waves = 256M x 32N; wave = 64M x 32N.
// B slab (16k x 32n) double-buffered in LDS via global_load_async_to_lds_b128.
__global__ __launch_bounds__(128) void k_score(const float* __restrict__ P,
                                               const float* __restrict__ Gt,
                                               float* __restrict__ out) {
    __shared__ __attribute__((aligned(16))) float sB[2][KCHUNK * 32]; // 2 x 2KB

    const int tid  = threadIdx.x;
    const int lane = tid & 31;
    const int wave = tid >> 5;                    // 0..3
    const int mb   = blockIdx.x * 256 + wave * 64;
    const int nb   = blockIdx.y * 32;             // 2560/32 = 80 blocks
    const int mrow = lane & 15;
    const int koff = (lane >> 4) * 2;
    const int ncol = lane & 15;

    // Staging geometry: thread tid copies 4 floats (one b128) per chunk.
    const int srow  = tid >> 3;                   // 0..15  (k-row in chunk)
    const int scol4 = (tid & 7) * 4;              // 0,4,...,28

    const v8f vzero = {0.f,0.f,0.f,0.f,0.f,0.f,0.f,0.f};
    v8f acc[4][2];
    const float* prow[4];
    #pragma unroll
    for (int t = 0; t < 4; ++t) {
        acc[t][0] = vzero; acc[t][1] = vzero;
        prow[t] = P + (size_t)(mb + t * 16 + mrow) * RANK_PAD;
    }

    // sB is the only __shared__ object -> LDS byte offset 0.
    #define ISSUE_CHUNK(c)                                                      \
        async_b128((unsigned)((((c) & 1) * (KCHUNK * 32) +                      \
                               srow * 32 + scol4) * sizeof(float)),             \
                   Gt + (size_t)((c) * KCHUNK + srow) * N_PAD + nb + scol4,     \
                   &sB[0][0])

    ISSUE_CHUNK(0);
    for (int c = 0; c < NCHUNK_S; ++c) {
        if (c + 1 < NCHUNK_S) { ISSUE_CHUNK(c + 1); wait_async1(); }
        else                  { wait_async(); }
        __syncthreads();                       // all 4 waves' pieces visible

        #pragma unroll
        for (int kl = 0; kl < KCHUNK; kl += 4) {
            const int r0 = (kl + koff) * 32 + ncol;
            v2f b0, b1;
            b0.x = sB[c & 1][r0];       b0.y = sB[c & 1][r0 + 32];
            b1.x = sB[c & 1][r0 + 16];  b1.y = sB[c & 1][r0 + 48];
            const int kb = c * KCHUNK + kl;
            #pragma unroll
            for (int t = 0; t < 4; ++t) {
                v2f a = *(const v2f*)(prow[t] + kb + koff);
                acc[t][0] = wmma_f32(a, b0, acc[t][0]);
                acc[t][1] = wmma_f32(a, b1, acc[t][1]);
            }
        }
        __syncthreads();                       // reads done before buffer reuse
    }
    #undef ISSUE_CHUNK

    #pragma unroll
    for (int t = 0; t < 4; ++t) {
        #pragma unroll
        for (int j = 0; j < 2; ++j) {
            int n = nb + j * 16 + ncol;
            if (n < N_TOTAL) {
                #pragma unroll
                for (int i = 0; i < 8; ++i) {
                    int m = mb + t * 16 + ((lane >> 4) << 3) + i;
                    out[(size_t)m * N_TOTAL + n] = acc[t][j][i];
                }
            }
        }
    }
}

extern "C" void kernel_launch(void* const* d_in, const int* in_sizes, int n_in,
                              void* d_out, int out_size, void* d_ws, size_t ws_size,
                              hipStream_t stream) {
    (void)in_sizes; (void)n_in; (void)out_size; (void)ws_size;
    const float* word  = (const float*)d_in[0];   // [512,32,400]
    const float* tag   = (const float*)d_in[1];   // [50,20]
    const float* T     = (const float*)d_in[2];   // [400,396]
    const float* U     = (const float*)d_in[3];   // [400,396]
    const float* V     = (const float*)d_in[4];   // [20,396]
    const float* W     = (const float*)d_in[5];   // [20,396]
    const float* first = (const float*)d_in[6];   // [1,400]
    float* out = (float*)d_out;                   // [512,32,50,50]

    // Workspace layout (offsets 256B-aligned, all 16B-aligned for b128 DMA):
    char* ws = (char*)d_ws;
    const size_t P_BYTES  = (size_t)M_TOTAL * RANK_PAD * sizeof(float);   // 26,214,400
    const size_t G_BYTES  = (size_t)RANK_PAD * N_PAD * sizeof(float);     //  4,096,000
    const size_t G23_B    = 79360;                                        // 50*396*4 padded
    const size_t UT_BYTES = (size_t)WEMD * RANK_PAD * sizeof(float);      //    640,000
    float* P  = (float*)ws;
    float* Gt = (float*)(ws + P_BYTES);
    float* g2 = (float*)(ws + P_BYTES + G_BYTES);
    float* g3 = (float*)(ws + P_BYTES + G_BYTES + G23_B);
    float* Up = (float*)(ws + P_BYTES + G_BYTES + 2 * G23_B);
    float* Tp = (float*)(ws + P_BYTES + G_BYTES + 2 * G23_B + UT_BYTES);

    {   // g2 = tag@V, g3 = tag@W  (tiny)
        int n = 2 * TAGSET * RANK;
        k_g23<<<(n + 255) / 256, 256, 0, stream>>>(tag, V, W, g2, g3);
    }
    {   // Gt[k][n] = g2[b,k]*g3[c,k], zero-padded to [400][2560]
        int n = RANK_PAD * N_PAD;
        k_buildG<<<(n + 255) / 256, 256, 0, stream>>>(g2, g3, Gt);
    }
    {   // Up/Tp = U/T zero-padded to [400][400]
        int n = 2 * WEMD * RANK_PAD;
        k_padUT<<<(n + 255) / 256, 256, 0, stream>>>(U, T, Up, Tp);
    }
    // P = (word@Up) * (neigh@Tp)   [16384 x 400]
    k_gates<<<dim3(M_TOTAL / 256, RANK_PAD / 16), 128, 0, stream>>>(word, first, Up, Tp, P);
    // score = P @ Gt               [16384 x 2500]
    k_score<<<dim3(M_TOTAL / 256, N_PAD / 32), 128, 0, stream>>>(P, Gt, out);
}